// res_block_77936476553474
// MI455X (gfx1250) — compile-verified
//
#include <hip/hip_runtime.h>

typedef __attribute__((ext_vector_type(2))) float v2f;
typedef __attribute__((ext_vector_type(8))) float v8f;

#define T1   4096
#define T2   8192
#define B_   32
#define C1   192      // in/out channels of block
#define HCH  384      // hidden channels
#define OFFC 72
#define K1   576      // 3 * 192
#define K2   1152     // 3 * 384
#define NT   4        // N tiles (of 16) per block
#define BTIME 64      // NT*16 time columns per block
#define LDS_S1 196    // 192 + 4 pad  -> stride%64 == 4, conflict-free b64
#define LDS_S2 388    // 384 + 4 pad  -> stride%64 == 4, conflict-free b64
#define BN_N ((float)((size_t)B_ * T2))

// dist<=2 neighbor bitmask over 24 skeleton joints (from PARENTS, Floyd-Warshall done offline)
__constant__ unsigned int NBR[24] = {
  0x00007Fu, 0x00009Fu, 0x00012Fu, 0x00024Fu, 0x000493u, 0x000925u,
  0x007249u, 0x000492u, 0x000924u, 0x03F248u, 0x000490u, 0x000920u,
  0x00F240u, 0x057240u, 0x0A7240u, 0x009200u, 0x152200u, 0x2A4200u,
  0x552000u, 0xAA4000u, 0x550000u, 0xAA0000u, 0x540000u, 0xA80000u };

// workspace layout (in floats)
enum : size_t {
  O_W1P   = 0,
  O_W2P   = O_W1P + (size_t)HCH * K1,          // 221184
  O_OFF1  = O_W2P + (size_t)C1 * K2,           // 221184
  O_OFF2  = O_OFF1 + (size_t)B_ * HCH,         // 12288
  O_STATS = O_OFF2 + (size_t)B_ * C1,          // 6144
  O_SCALE = O_STATS + 1152,
  O_H1    = O_SCALE + 1152,
  O_H2    = O_H1 + (size_t)B_ * HCH * T2,
};

__device__ __forceinline__ float upsample_val(const float* __restrict__ xc, int t2) {
  // torch Upsample(scale=2, linear, align_corners=False)
  int t = t2 >> 1;
  if (t2 == 0)            return xc[0];
  if ((t2 & 1) == 0)      return 0.25f * xc[t - 1] + 0.75f * xc[t];
  if (t == T1 - 1)        return xc[T1 - 1];
  return 0.75f * xc[t] + 0.25f * xc[t + 1];
}

__global__ void zero_stats_k(float* __restrict__ p) {
  int i = blockIdx.x * blockDim.x + threadIdx.x;
  if (i < 2304) p[i] = 0.f;
}

// masked conv1 weights -> [m][k], k = tap*C1 + cin
__global__ void prep_w1_k(const float* __restrict__ w, float* __restrict__ w1p) {
  int q = blockIdx.x * blockDim.x + threadIdx.x;
  if (q >= HCH * K1) return;
  int m = q / K1, r = q - m * K1;
  int tap = r / C1, cin = r - tap * C1;
  bool on = (NBR[m >> 4] >> (cin >> 3)) & 1u;
  w1p[q] = on ? w[(m * C1 + cin) * 3 + tap] : 0.f;
}

__global__ void prep_w2_k(const float* __restrict__ w, float* __restrict__ w2p) {
  int q = blockIdx.x * blockDim.x + threadIdx.x;
  if (q >= C1 * K2) return;
  int m = q / K2, r = q - m * K2;
  int tap = r / HCH, cin = r - tap * HCH;
  bool on = (NBR[m >> 3] >> (cin >> 4)) & 1u;
  w2p[q] = on ? w[(m * HCH + cin) * 3 + tap] : 0.f;
}

// per-(batch,channel) offset bias: (offset @ (W*mask)^T + b) / 100
__global__ void prep_off_k(const float* __restrict__ offset,
                           const float* __restrict__ o1w, const float* __restrict__ o1b,
                           const float* __restrict__ o2w, const float* __restrict__ o2b,
                           float* __restrict__ off1, float* __restrict__ off2) {
  int q = blockIdx.x * blockDim.x + threadIdx.x;
  if (q < B_ * HCH) {
    int b = q / HCH, m = q - b * HCH;
    unsigned nb = NBR[m >> 4];
    float acc = o1b[m];
    for (int c = 0; c < OFFC; ++c)
      if ((nb >> (c / 3)) & 1u) acc += offset[b * OFFC + c] * o1w[m * OFFC + c];
    off1[q] = acc * 0.01f;
  } else if (q < B_ * HCH + B_ * C1) {
    int q2 = q - B_ * HCH;
    int b = q2 / C1, m = q2 - b * C1;
    unsigned nb = NBR[m >> 3];
    float acc = o2b[m];
    for (int c = 0; c < OFFC; ++c)
      if ((nb >> (c / 3)) & 1u) acc += offset[b * OFFC + c] * o2w[m * OFFC + c];
    off2[q2] = acc * 0.01f;
  }
}

// conv1: h1 = conv(upsample(x), w1_masked) + b1 + off1 ; accumulate BN1 sums
__global__ __launch_bounds__(256) void conv1_k(
    const float* __restrict__ x, const float* __restrict__ w1p,
    const float* __restrict__ b1, const float* __restrict__ off1,
    float* __restrict__ h1, float* __restrict__ bnsum, float* __restrict__ bnssq) {
  extern __shared__ float U[];                 // [66][LDS_S1], time-major
  const int b = blockIdx.y;
  const int t2base = blockIdx.x * BTIME;
  const int tid = threadIdx.x;
  const float* xb = x + (size_t)b * C1 * T1;

  for (int idx = tid; idx < 66 * C1; idx += 256) {
    int c = idx / 66, tt = idx - c * 66;
    int t2 = t2base + tt - 1;                  // halo of 1 for k=3, pad=1 (zeros)
    float v = 0.f;
    if (t2 >= 0 && t2 < T2) v = upsample_val(xb + (size_t)c * T1, t2);
    U[tt * LDS_S1 + c] = v;
  }
  __syncthreads();

  const int lane = tid & 31, wave = tid >> 5;  // 8 waves
  const int nlo = lane & 15;
  const int khalf = (lane >> 4) << 1;          // 0 or 2 (A/B K sub-lanes)

  v8f acc[3][NT];
  for (int i = 0; i < 3; ++i)
    for (int j = 0; j < NT; ++j)
      for (int e = 0; e < 8; ++e) acc[i][j][e] = 0.f;

  for (int k0 = 0; k0 < K1; k0 += 4) {
    const int tap = k0 / C1;
    const int cin = (k0 - tap * C1) + khalf;
    v2f bf[NT];
#pragma unroll
    for (int j = 0; j < NT; ++j)
      bf[j] = *(const v2f*)&U[(j * 16 + nlo + tap) * LDS_S1 + cin];
#pragma unroll
    for (int i = 0; i < 3; ++i) {
      const int m = (wave + 8 * i) * 16 + nlo;
      v2f af = *(const v2f*)&w1p[(size_t)m * K1 + k0 + khalf];
#pragma unroll
      for (int j = 0; j < NT; ++j)
        acc[i][j] = __builtin_amdgcn_wmma_f32_16x16x4_f32(
            false, af, false, bf[j], (short)0, acc[i][j], false, false);
    }
  }

  const int mhalf = (lane >> 4) << 3;
  for (int i = 0; i < 3; ++i) {
#pragma unroll
    for (int v = 0; v < 8; ++v) {
      const int m = (wave + 8 * i) * 16 + mhalf + v;
      const float add = b1[m] + off1[b * HCH + m];
      float ps = 0.f, pq = 0.f;
#pragma unroll
      for (int j = 0; j < NT; ++j) {
        float val = acc[i][j][v] + add;
        h1[((size_t)b * HCH + m) * T2 + t2base + j * 16 + nlo] = val;
        ps += val; pq += val * val;
      }
      for (int s = 1; s < 16; s <<= 1) {       // reduce within each 16-lane half
        ps += __shfl_xor(ps, s, 32);
        pq += __shfl_xor(pq, s, 32);
      }
      if (nlo == 0) { atomicAdd(&bnsum[m], ps); atomicAdd(&bnssq[m], pq); }
    }
  }
}

__global__ void bnfin_k(const float* __restrict__ sum, const float* __restrict__ ssq,
                        const float* __restrict__ g, const float* __restrict__ bb,
                        float* __restrict__ s, float* __restrict__ t, int nch) {
  int c = threadIdx.x;
  if (c < nch) {
    float m = sum[c] / BN_N;
    float v = ssq[c] / BN_N - m * m;
    float sc = g[c] * rsqrtf(v + 1e-5f);
    s[c] = sc;
    t[c] = bb[c] - m * sc;
  }
}

// conv2: stage relu(bn1(h1)) tile in LDS, GEMM with masked w2; accumulate BN2 sums
__global__ __launch_bounds__(384) void conv2_k(
    const float* __restrict__ h1, const float* __restrict__ w2p,
    const float* __restrict__ s1, const float* __restrict__ t1,
    const float* __restrict__ b2, const float* __restrict__ off2,
    float* __restrict__ h2, float* __restrict__ bnsum, float* __restrict__ bnssq) {
  extern __shared__ float Hs[];                // [66][LDS_S2]
  const int b = blockIdx.y;
  const int t2base = blockIdx.x * BTIME;
  const int tid = threadIdx.x;

  for (int idx = tid; idx < 66 * HCH; idx += 384) {
    int c = idx / 66, tt = idx - c * 66;
    int t2 = t2base + tt - 1;
    float v = 0.f;
    if (t2 >= 0 && t2 < T2) {
      float hv = h1[((size_t)b * HCH + c) * T2 + t2];
      v = fmaxf(fmaf(s1[c], hv, t1[c]), 0.f);  // fused BN1 + ReLU
    }
    Hs[tt * LDS_S2 + c] = v;
  }
  __syncthreads();

  const int lane = tid & 31, wave = tid >> 5;  // 12 waves -> 12 M tiles (192 ch)
  const int nlo = lane & 15;
  const int khalf = (lane >> 4) << 1;

  v8f acc[NT];
  for (int j = 0; j < NT; ++j)
    for (int e = 0; e < 8; ++e) acc[j][e] = 0.f;

  for (int k0 = 0; k0 < K2; k0 += 4) {
    const int tap = k0 / HCH;
    const int cin = (k0 - tap * HCH) + khalf;
    const int m = wave * 16 + nlo;
    v2f af = *(const v2f*)&w2p[(size_t)m * K2 + k0 + khalf];
#pragma unroll
    for (int j = 0; j < NT; ++j) {
      v2f bf = *(const v2f*)&Hs[(j * 16 + nlo + tap) * LDS_S2 + cin];
      acc[j] = __builtin_amdgcn_wmma_f32_16x16x4_f32(
          false, af, false, bf, (short)0, acc[j], false, false);
    }
  }

  const int mhalf = (lane >> 4) << 3;
#pragma unroll
  for (int v = 0; v < 8; ++v) {
    const int m = wave * 16 + mhalf + v;
    const float add = b2[m] + off2[b * C1 + m];
    float ps = 0.f, pq = 0.f;
#pragma unroll
    for (int j = 0; j < NT; ++j) {
      float val = acc[j][v] + add;
      h2[((size_t)b * C1 + m) * T2 + t2base + j * 16 + nlo] = val;
      ps += val; pq += val * val;
    }
    for (int s = 1; s < 16; s <<= 1) {
      ps += __shfl_xor(ps, s, 32);
      pq += __shfl_xor(pq, s, 32);
    }
    if (nlo == 0) { atomicAdd(&bnsum[m], ps); atomicAdd(&bnssq[m], pq); }
  }
}

// out = relu(bn2(h2) + upsample(x))  (residual recomputed on the fly)
__global__ __launch_bounds__(256) void final_k(
    const float* __restrict__ h2, const float* __restrict__ x,
    const float* __restrict__ s2, const float* __restrict__ t2c,
    float* __restrict__ out) {
  size_t idx = (size_t)blockIdx.x * 256 + threadIdx.x;
  int t2 = (int)(idx % T2);
  size_t rest = idx / T2;
  int c = (int)(rest % C1);
  int b = (int)(rest / C1);
  const float* xc = x + ((size_t)b * C1 + c) * T1;
  float u = upsample_val(xc, t2);
  float val = fmaf(s2[c], h2[idx], t2c[c]) + u;
  out[idx] = fmaxf(val, 0.f);
}

extern "C" void kernel_launch(void* const* d_in, const int* in_sizes, int n_in,
                              void* d_out, int out_size, void* d_ws, size_t ws_size,
                              hipStream_t stream) {
  const float* x       = (const float*)d_in[0];
  const float* offset  = (const float*)d_in[1];
  const float* conv1_w = (const float*)d_in[2];
  const float* conv1_b = (const float*)d_in[3];
  const float* off1_w  = (const float*)d_in[4];
  const float* off1_b  = (const float*)d_in[5];
  const float* bn1_g   = (const float*)d_in[6];
  const float* bn1_b   = (const float*)d_in[7];
  const float* conv2_w = (const float*)d_in[8];
  const float* conv2_b = (const float*)d_in[9];
  const float* off2_w  = (const float*)d_in[10];
  const float* off2_b  = (const float*)d_in[11];
  const float* bn2_g   = (const float*)d_in[12];
  const float* bn2_b   = (const float*)d_in[13];
  float* out = (float*)d_out;

  float* ws   = (float*)d_ws;
  float* w1p  = ws + O_W1P;
  float* w2p  = ws + O_W2P;
  float* off1 = ws + O_OFF1;
  float* off2 = ws + O_OFF2;
  float* sum1 = ws + O_STATS;        // 384
  float* ssq1 = ws + O_STATS + 384;  // 384
  float* sum2 = ws + O_STATS + 768;  // 192
  float* ssq2 = ws + O_STATS + 960;  // 192
  float* s1   = ws + O_SCALE;
  float* t1   = ws + O_SCALE + 384;
  float* s2   = ws + O_SCALE + 768;
  float* t2v  = ws + O_SCALE + 960;
  float* h1   = ws + O_H1;
  float* h2   = ws + O_H2;

  zero_stats_k<<<9, 256, 0, stream>>>(ws + O_STATS);
  prep_w1_k<<<(HCH * K1 + 255) / 256, 256, 0, stream>>>(conv1_w, w1p);
  prep_w2_k<<<(C1 * K2 + 255) / 256, 256, 0, stream>>>(conv2_w, w2p);
  prep_off_k<<<(B_ * (HCH + C1) + 255) / 256, 256, 0, stream>>>(
      offset, off1_w, off1_b, off2_w, off2_b, off1, off2);

  dim3 g(T2 / BTIME, B_);
  conv1_k<<<g, 256, 66 * LDS_S1 * sizeof(float), stream>>>(
      x, w1p, conv1_b, off1, h1, sum1, ssq1);
  bnfin_k<<<1, HCH, 0, stream>>>(sum1, ssq1, bn1_g, bn1_b, s1, t1, HCH);
  conv2_k<<<g, 384, 66 * LDS_S2 * sizeof(float), stream>>>(
      h1, w2p, s1, t1, conv2_b, off2, h2, sum2, ssq2);
  bnfin_k<<<1, C1, 0, stream>>>(sum2, ssq2, bn2_g, bn2_b, s2, t2v, C1);
  final_k<<<(int)(((size_t)B_ * C1 * T2) / 256), 256, 0, stream>>>(h2, x, s2, t2v, out);
}